// HeatmapOffsetmapLoss_26362509263028
// MI455X (gfx1250) — compile-verified
//
#include <hip/hip_runtime.h>

typedef __attribute__((ext_vector_type(2))) float v2f;
typedef __attribute__((ext_vector_type(8))) float v8f;

#define HH 512
#define WW 512
#define PLANE (HH * WW)          // 262144 elements per channel plane
#define NLM 19                   // landmarks per batch
#define NPAIR 76                 // 4 batches * 19 landmarks
#define BPP 32                   // blocks per (b, n) pair
#define THREADS 256              // 8 wave32 per block
#define ITERS 8                  // (PLANE/BPP)/4/THREADS = 8192/4/256

// Wave32 sum via V_WMMA_F32_16X16X4_F32 with an all-ones B matrix.
// A layout (documented): lanes 0-15 VGPR0 = A[m][0], lanes 16-31 VGPR0 = A[m][2];
// VGPR1 (K=1/K=3) set to 0.  With B = ones: D[m][n] = acc[m] + acc[m+16].
// C/D layout: lane L<16, VGPR v -> (M=v, N=L); lane L>=16 -> (M=v+8).
// So summing the 8 D registers in-lane gives half-wave sums replicated across
// each 16-lane half; one xor-16 shuffle completes the full 32-lane sum.
__device__ __forceinline__ float wave_sum32(float v) {
  v2f a;     a[0] = v;    a[1] = 0.0f;
  v2f bones; bones[0] = 1.0f; bones[1] = 1.0f;
  v8f c = {};
  v8f d = __builtin_amdgcn_wmma_f32_16x16x4_f32(
      /*neg_a=*/false, a, /*neg_b=*/false, bones,
      /*c_mod=*/(short)0, c, /*reuse_a=*/false, /*reuse_b=*/false);
  float t = ((d[0] + d[1]) + (d[2] + d[3])) + ((d[4] + d[5]) + (d[6] + d[7]));
  t += __shfl_xor(t, 16, 32);
  return t;  // all lanes hold the wave-wide sum
}

__global__ void __launch_bounds__(THREADS)
heatmap_loss_partials(const float* __restrict__ fm, const float* __restrict__ lm,
                      float* __restrict__ part) {
  const int bid  = blockIdx.x;
  const int pair = bid / BPP;     // b*19 + n
  const int blk  = bid % BPP;
  const int b = pair / NLM;
  const int n = pair % NLM;

  // xy = int32(landmark * [H, W]); truncation matches astype(int32)
  const int x = (int)(lm[pair * 2 + 0] * 512.0f);
  const int y = (int)(lm[pair * 2 + 1] * 512.0f);

  const float4* lg = (const float4*)(fm + (size_t)(b * 57 + n) * PLANE);
  const float4* px = (const float4*)(fm + (size_t)(b * 57 + NLM + n) * PLANE);
  const float4* py = (const float4*)(fm + (size_t)(b * 57 + 2 * NLM + n) * PLANE);

  const int base4 = blk * (PLANE / BPP / 4);  // float4 units, 2048 per block

  float accB = 0.0f, accC = 0.0f, accX = 0.0f, accY = 0.0f;

#pragma unroll
  for (int it = 0; it < ITERS; ++it) {
    const int e4 = base4 + it * THREADS + threadIdx.x;
    const float4 l4 = lg[e4];
    const float4 x4 = px[e4];
    const float4 y4 = py[e4];
    const int e = e4 << 2;          // flat element index in plane
    const int i = e >> 9;           // row (H index)
    const int j = e & (WW - 1);     // col (W index), float4 stays within a row
    const float fdi = (float)(i - x);
    const float di2 = fdi * fdi;
    const float lv[4] = {l4.x, l4.y, l4.z, l4.w};
    const float xv[4] = {x4.x, x4.y, x4.z, x4.w};
    const float yv[4] = {y4.x, y4.y, y4.z, y4.w};
#pragma unroll
    for (int k = 0; k < 4; ++k) {
      const float fdj = (float)(j + k - y);
      // heatmap: dist <= 40  <=>  di^2 + dj^2 <= 1600 (avoid sqrt)
      const float h = (di2 + fdj * fdj <= 1600.0f) ? 1.0f : 0.0f;
      const float l = lv[k];
      // stable BCE-with-logits: max(l,0) - l*h + log1p(exp(-|l|))
      accB += fmaxf(l, 0.0f) - l * h + __logf(1.0f + __expf(-fabsf(l)));
      accC += h;
      // offmap_x = -di/40  =>  |pred_x - offmap_x| = |pred_x + di/40|
      accX += fabsf(xv[k] + fdi * 0.025f) * h;
      accY += fabsf(yv[k] + fdj * 0.025f) * h;
    }
  }

  // Wave reduction via WMMA (EXEC is all-ones here: no divergence above).
  const float wb = wave_sum32(accB);
  const float wc = wave_sum32(accC);
  const float wx = wave_sum32(accX);
  const float wy = wave_sum32(accY);

  __shared__ float red[THREADS / 32][4];
  const int wave = threadIdx.x >> 5;
  const int lane = threadIdx.x & 31;
  if (lane == 0) {
    red[wave][0] = wb; red[wave][1] = wc; red[wave][2] = wx; red[wave][3] = wy;
  }
  __syncthreads();
  if (threadIdx.x < 4) {
    float s = 0.0f;
#pragma unroll
    for (int w = 0; w < THREADS / 32; ++w) s += red[w][threadIdx.x];
    part[(size_t)bid * 4 + threadIdx.x] = s;  // deterministic, no atomics
  }
}

__global__ void __launch_bounds__(128)
heatmap_loss_finalize(const float* __restrict__ part, float* __restrict__ out) {
  __shared__ float buf[128];
  const int t = threadIdx.x;
  float loss = 0.0f;
  if (t < NPAIR) {
    float sb = 0.0f, sc = 0.0f, sx = 0.0f, sy = 0.0f;
    for (int blk = 0; blk < BPP; ++blk) {
      const float* p = part + (size_t)(t * BPP + blk) * 4;
      sb += p[0]; sc += p[1]; sx += p[2]; sy += p[3];
    }
    // 2*mean(bce) + l1x/cnt + l1y/cnt
    loss = 2.0f * (sb * (1.0f / (float)PLANE)) + (sx + sy) / sc;
  }
  buf[t] = loss;
  __syncthreads();
  for (int s = 64; s > 0; s >>= 1) {
    if (t < s) buf[t] += buf[t + s];
    __syncthreads();
  }
  if (t == 0) out[0] = buf[0] * (1.0f / (float)NPAIR);
}

extern "C" void kernel_launch(void* const* d_in, const int* in_sizes, int n_in,
                              void* d_out, int out_size, void* d_ws, size_t ws_size,
                              hipStream_t stream) {
  const float* fm = (const float*)d_in[0];  // (4, 57, 512, 512) f32
  const float* lm = (const float*)d_in[1];  // (4, 19, 2) f32
  float* part = (float*)d_ws;               // 2432 * 4 floats = 38 KB scratch

  heatmap_loss_partials<<<NPAIR * BPP, THREADS, 0, stream>>>(fm, lm, part);
  heatmap_loss_finalize<<<1, 128, 0, stream>>>(part, (float*)d_out);
}